// LocalPlasticityNet_30863634989577
// MI455X (gfx1250) — compile-verified
//
#include <hip/hip_runtime.h>
#include <hip/hip_bf16.h>

// ---------------------------------------------------------------------------
// LocalPlasticityNet on MI455X (gfx1250, wave32)
//   1) per-row real FFT(20000) via packed complex Stockham FFT(10000) in LDS
//      (templated constant-parameter stages; 1 sincos per radix-5 butterfly)
//   2) three fused Hebbian layers using V_WMMA_F32_16X16X4_F32, with the
//      activation slab staged into LDS via async global->LDS (double buffered)
//   3) head dot product fused into the last layer
// ---------------------------------------------------------------------------

typedef __attribute__((ext_vector_type(2))) float v2f;
typedef __attribute__((ext_vector_type(8))) float v8f;
typedef int async_v4i __attribute__((vector_size(16)));   // builtin's pointee type

#define FFT_N 10000          // complex FFT size (packs 20000 real samples)
#define FFT_THREADS 256
#define BATCH 1024
#define NFREQ 10001
#define K0PAD 10240          // 10001 padded to a multiple of 256 (K-chunking)

// ---------------- async global->LDS helpers (CDNA5) ----------------
#if __has_builtin(__builtin_amdgcn_global_load_async_to_lds_b128) && \
    __has_builtin(__builtin_amdgcn_s_wait_asynccnt)
#define USE_ASYNC_LDS 1
#else
#define USE_ASYNC_LDS 0
#endif

__device__ __forceinline__ void copy16_g2l(const float* g, float* l)
{
#if USE_ASYNC_LDS
    __builtin_amdgcn_global_load_async_to_lds_b128(
        (async_v4i*)(void*)g, (async_v4i*)(void*)l, 0, 0);
#else
    *(float4*)l = *(const float4*)g;
#endif
}

template <int N>
__device__ __forceinline__ void async_wait()
{
#if USE_ASYNC_LDS
    __builtin_amdgcn_s_wait_asynccnt(N);
#endif
}

// ---------------- complex helpers ----------------
__device__ __forceinline__ float2 cadd(float2 a, float2 b) { return make_float2(a.x + b.x, a.y + b.y); }
__device__ __forceinline__ float2 csub(float2 a, float2 b) { return make_float2(a.x - b.x, a.y - b.y); }
__device__ __forceinline__ float2 cmul(float2 a, float2 b) {
    return make_float2(a.x * b.x - a.y * b.y, a.x * b.y + a.y * b.x);
}

// ---------------------------------------------------------------------------
// One Stockham DIF stage with compile-time radix R, length N, stride S.
// All index div/mod are by constants (strength-reduced); twiddles for radix-5
// use one sincos + complex powers.
// ---------------------------------------------------------------------------
template <int R, int N, int S>
__device__ __forceinline__ void fft_stage(const float2* __restrict__ X,
                                          float2* __restrict__ Y, int tid)
{
    constexpr int M = N / R;
    constexpr int NB = M * S;
    constexpr float NEG2PI_N = -6.28318530717958647692f / (float)N;
    const float2 W5C[5] = {
        make_float2( 1.0f,                  0.0f),
        make_float2( 0.30901699437494742f, -0.95105651629515357f),
        make_float2(-0.80901699437494745f, -0.58778525229247312f),
        make_float2(-0.80901699437494745f,  0.58778525229247312f),
        make_float2( 0.30901699437494742f,  0.95105651629515357f)};

    for (int id = tid; id < NB; id += FFT_THREADS) {
        const int q = id % S;
        const int p = id / S;
        if constexpr (R == 5) {
            float2 a0 = X[q + S * p];
            float2 a1 = X[q + S * (p + M)];
            float2 a2 = X[q + S * (p + 2 * M)];
            float2 a3 = X[q + S * (p + 3 * M)];
            float2 a4 = X[q + S * (p + 4 * M)];
            float2 bs[5];
            #pragma unroll
            for (int u = 0; u < 5; ++u) {
                float2 acc = a0;
                acc = cadd(acc, cmul(a1, W5C[(1 * u) % 5]));
                acc = cadd(acc, cmul(a2, W5C[(2 * u) % 5]));
                acc = cadd(acc, cmul(a3, W5C[(3 * u) % 5]));
                acc = cadd(acc, cmul(a4, W5C[(4 * u) % 5]));
                bs[u] = acc;
            }
            float sn, cs;
            sincosf(NEG2PI_N * (float)p, &sn, &cs);
            const float2 w1 = make_float2(cs, sn);
            const float2 w2 = cmul(w1, w1);
            const float2 w3 = cmul(w2, w1);
            const float2 w4 = cmul(w2, w2);
            Y[q + S * (5 * p + 0)] = bs[0];
            Y[q + S * (5 * p + 1)] = cmul(bs[1], w1);
            Y[q + S * (5 * p + 2)] = cmul(bs[2], w2);
            Y[q + S * (5 * p + 3)] = cmul(bs[3], w3);
            Y[q + S * (5 * p + 4)] = cmul(bs[4], w4);
        } else {
            float2 a0 = X[q + S * p];
            float2 a1 = X[q + S * (p + M)];
            float sn, cs;
            sincosf(NEG2PI_N * (float)p, &sn, &cs);
            Y[q + S * (2 * p)]     = cadd(a0, a1);
            Y[q + S * (2 * p + 1)] = cmul(csub(a0, a1), make_float2(cs, sn));
        }
    }
}

// ---------------------------------------------------------------------------
// FFT front-end: one workgroup per batch row.
//   z[n] = x[2n] + i x[2n+1]; Z = FFT_10000(z); rfft unpack; h = log1p(|X_k|)
//   h[:,0] = 0 exactly (mean-centering only affects the DC bin).
// Also emits rnorm0[b] = 1/(||h_b|| + 1e-8) for the Hebbian trace term.
// ---------------------------------------------------------------------------
__global__ __launch_bounds__(FFT_THREADS) void fft_front_kernel(
    const float* __restrict__ x, float* __restrict__ h0, float* __restrict__ rnorm0)
{
    extern __shared__ float2 smem[];     // 2 * FFT_N float2 = 160000 bytes
    __shared__ float s_red[FFT_THREADS];
    float2* bufA = smem;
    float2* bufB = smem + FFT_N;

    const int b = blockIdx.x;
    const int tid = threadIdx.x;
    const float* xr = x + (size_t)b * 20000;

    for (int n = tid; n < FFT_N; n += FFT_THREADS)
        bufA[n] = *(const float2*)(xr + 2 * n);
    __syncthreads();

    fft_stage<5, 10000, 1>(bufA, bufB, tid);  __syncthreads();
    fft_stage<5, 2000, 5>(bufB, bufA, tid);   __syncthreads();
    fft_stage<5, 400, 25>(bufA, bufB, tid);   __syncthreads();
    fft_stage<5, 80, 125>(bufB, bufA, tid);   __syncthreads();
    fft_stage<2, 16, 625>(bufA, bufB, tid);   __syncthreads();
    fft_stage<2, 8, 1250>(bufB, bufA, tid);   __syncthreads();
    fft_stage<2, 4, 2500>(bufA, bufB, tid);   __syncthreads();
    fft_stage<2, 2, 5000>(bufB, bufA, tid);   __syncthreads();
    // natural-order result now in bufA

    float local = 0.0f;
    float* hrow = h0 + (size_t)b * K0PAD;
    for (int k = tid; k < K0PAD; k += FFT_THREADS) {
        float val = 0.0f;
        if (k > 0 && k <= FFT_N) {
            const int i1 = (k == FFT_N) ? 0 : k;
            const int i2 = (FFT_N - k) % FFT_N;
            float2 z1 = bufA[i1];
            float2 z2 = bufA[i2];
            // E = (z1 + conj(z2))/2 ; O = -i*(z1 - conj(z2))/2
            float Ere = 0.5f * (z1.x + z2.x);
            float Eim = 0.5f * (z1.y - z2.y);
            float Dre = z1.x - z2.x;
            float Dim = z1.y + z2.y;
            float Ore = 0.5f * Dim;
            float Oim = -0.5f * Dre;
            float ang = -3.14159265358979323846f * (float)k / (float)FFT_N; // -2*pi*k/20000
            float sn, cs;
            sincosf(ang, &sn, &cs);
            float Xre = Ere + cs * Ore - sn * Oim;
            float Xim = Eim + cs * Oim + sn * Ore;
            val = log1pf(sqrtf(Xre * Xre + Xim * Xim));
        }
        hrow[k] = val;                 // k==0 and pad columns -> 0
        local += val * val;
    }
    s_red[tid] = local;
    __syncthreads();
    if (tid == 0) {                    // deterministic serial tail
        float ss = 0.0f;
        for (int i = 0; i < FFT_THREADS; ++i) ss += s_red[i];
        rnorm0[b] = 1.0f / (sqrtf(ss) + 1e-8f);
    }
}

// ---------------------------------------------------------------------------
// Weight pre-pack: Wp = W zero-padded to Kp; Ap = sigmoid(alpha) zero-padded.
// ---------------------------------------------------------------------------
__global__ void prepack_kernel(const float* __restrict__ W, const float* __restrict__ al,
                               float* __restrict__ Wp, float* __restrict__ Ap,
                               int K, int Kp, int total)
{
    int idx = blockIdx.x * blockDim.x + threadIdx.x;
    if (idx >= total) return;
    int i = idx / Kp, j = idx - i * Kp;
    if (j < K) {
        Wp[idx] = W[(size_t)i * K + j];
        float a = al[(size_t)i * K + j];
        Ap[idx] = 1.0f / (1.0f + expf(-a));
    } else {
        Wp[idx] = 0.0f;
        Ap[idx] = 0.0f;
    }
}

// ---------------------------------------------------------------------------
// Fused Hebbian layer (f32 WMMA):
//   Activation slab (16 batch rows x KC columns) staged in LDS by async
//   global->LDS copies, double buffered; weights streamed from L2.
//   Per wave: 16x16 tile, two V_WMMA_F32_16X16X4_F32 per K-step
//   (slow path h.W^T and trace path h^2.sigma(alpha)^T; 1/||h|| folded into
//   the epilogue since it is constant per batch row).
//   Workgroup: 16 batch rows x all H columns -> LayerNorm/GELU/row-norm in LDS.
// ---------------------------------------------------------------------------
template <int INP, int H, bool HEAD>
__global__ __launch_bounds__(2 * H) void hebb_layer_kernel(
    const float* __restrict__ hin, const float* __restrict__ rn_in,
    const float* __restrict__ Wp, const float* __restrict__ Ap,
    const float* __restrict__ bias, const float* __restrict__ eta_p,
    const float* __restrict__ gamma, const float* __restrict__ beta,
    float* __restrict__ hout, float* __restrict__ rn_out,
    const float* __restrict__ head_w, const float* __restrict__ head_b,
    float* __restrict__ dout)
{
    constexpr int KC   = (INP > 256) ? 256 : INP;  // K chunk per LDS slab
    constexpr int NCH  = INP / KC;                 // 40 / 1 / 1
    constexpr int LDR  = KC + 4;                   // slab row stride: 4 banks/row
    constexpr int SLAB = 16 * LDR;                 // floats per buffer
    constexpr int NBUF = (NCH > 1) ? 2 : 1;
    constexpr int IOV  = (16 * (KC / 4)) / (2 * H); // b128 copies per thread/chunk

    __shared__ __align__(16) float s_slab[NBUF * SLAB];
    __shared__ float s_pre[16 * H];
    __shared__ float s_p0[2 * H];
    __shared__ float s_p1[2 * H];
    __shared__ float s_mu[16];
    __shared__ float s_rstd[16];

    const int tid   = threadIdx.x;
    const int wave  = tid >> 5;
    const int lane  = tid & 31;
    const int n0    = wave * 16;          // column tile of this wave
    const int bt0   = blockIdx.x * 16;    // batch tile of this workgroup
    const int rA    = lane & 15;          // per-lane matrix row (M for A, N for B)
    const int khalf = (lane >> 4) * 2;    // K sub-offset per ISA 16x4 layout

    const float* wptr = Wp + (size_t)(n0 + rA) * INP + khalf;
    const float* pptr = Ap + (size_t)(n0 + rA) * INP + khalf;

    // issue async copies for chunk c into buffer c&1 (coalesced: 64 threads/row)
    auto issue = [&](int c) {
        const int k0 = c * KC;
        float* dst = s_slab + (c & 1) * SLAB;
        #pragma unroll
        for (int t = 0; t < IOV; ++t) {
            const int i = tid + t * (2 * H);
            const int r = i / (KC / 4);
            const int j = i - r * (KC / 4);
            copy16_g2l(hin + (size_t)(bt0 + r) * INP + k0 + 4 * j,
                       dst + r * LDR + 4 * j);
        }
    };

    v8f accW = {0.f, 0.f, 0.f, 0.f, 0.f, 0.f, 0.f, 0.f};
    v8f accP = {0.f, 0.f, 0.f, 0.f, 0.f, 0.f, 0.f, 0.f};

    issue(0);
    for (int c = 0; c < NCH; ++c) {
        if (c + 1 < NCH) {
            issue(c + 1);
            async_wait<IOV>();            // chunk c landed; c+1 in flight
        } else {
            async_wait<0>();              // drain
        }
        __syncthreads();

        const float* abuf = s_slab + (c & 1) * SLAB + rA * LDR + khalf;
        const int kg = c * KC;
        __builtin_prefetch(wptr + kg + 2 * KC, 0, 1);   // global_prefetch_b8
        __builtin_prefetch(pptr + kg + 2 * KC, 0, 1);
        #pragma unroll 4
        for (int kk = 0; kk < KC; kk += 4) {
            v2f a = *(const v2f*)(abuf + kk);            // ds_load_b64, conflict-free
            v2f w = *(const v2f*)(wptr + kg + kk);
            v2f p = *(const v2f*)(pptr + kg + kk);
            v2f c2 = a * a;                              // rn folded into epilogue
            accW = __builtin_amdgcn_wmma_f32_16x16x4_f32(false, a,  false, w, (short)0, accW, false, false);
            accP = __builtin_amdgcn_wmma_f32_16x16x4_f32(false, c2, false, p, (short)0, accP, false, false);
        }
        __syncthreads();                  // slab reuse safe next iteration
    }

    // epilogue phase 1: y_slow + eta*tanh(y_slow)*(rn_row*plastic) -> LDS
    const float eta = eta_p[0];
    const int cn    = n0 + (lane & 15);   // output column of this lane
    const float bb  = bias[cn];
    const int mbase = (lane >> 4) * 8;    // C/D layout: lanes>=16 hold rows 8..15
    #pragma unroll
    for (int v = 0; v < 8; ++v) {
        const float rnm = rn_in[bt0 + mbase + v];
        float y = accW[v] + bb;
        float pre = y + eta * tanhf(y) * (accP[v] * rnm);
        s_pre[(mbase + v) * H + cn] = pre;
    }
    __syncthreads();

    // phase 2: LayerNorm stats (deterministic two-level reduction)
    const int base = tid * 8;             // 8 | H -> span stays within one row
    const int row  = base / H;
    float sum = 0.0f, sq = 0.0f;
    float vals[8];
    #pragma unroll
    for (int e = 0; e < 8; ++e) {
        float t = s_pre[base + e];
        vals[e] = t;
        sum += t;
        sq += t * t;
    }
    s_p0[tid] = sum;
    s_p1[tid] = sq;
    __syncthreads();
    if (tid < 16) {
        constexpr int PPR = H / 8;        // partials per row (contiguous)
        float S = 0.0f, Q = 0.0f;
        for (int i = 0; i < PPR; ++i) {
            S += s_p0[tid * PPR + i];
            Q += s_p1[tid * PPR + i];
        }
        float mu = S / (float)H;
        float var = Q / (float)H - mu * mu;
        s_mu[tid] = mu;
        s_rstd[tid] = rsqrtf(var + 1e-5f);
    }
    __syncthreads();

    // LN + exact GELU; accumulate next-layer row norm / head dot
    const float mu = s_mu[row], rstd = s_rstd[row];
    const int col0 = base - row * H;
    float g2 = 0.0f, hd = 0.0f;
    #pragma unroll
    for (int e = 0; e < 8; ++e) {
        int col = col0 + e;
        float ln = (vals[e] - mu) * rstd * gamma[col] + beta[col];
        float gel = 0.5f * ln * (1.0f + erff(ln * 0.70710678118654752f));
        if constexpr (!HEAD) hout[(size_t)(bt0 + row) * H + col] = gel;
        g2 += gel * gel;
        if constexpr (HEAD) hd += gel * head_w[col];
    }
    s_p0[tid] = g2;
    s_p1[tid] = hd;
    __syncthreads();
    if (tid < 16) {
        constexpr int PPR = H / 8;
        float G = 0.0f, Hs = 0.0f;
        for (int i = 0; i < PPR; ++i) {
            G  += s_p0[tid * PPR + i];
            Hs += s_p1[tid * PPR + i];
        }
        if constexpr (HEAD) dout[bt0 + tid] = Hs + head_b[0];
        else                rn_out[bt0 + tid] = 1.0f / (sqrtf(G) + 1e-8f);
    }
}

// ---------------------------------------------------------------------------
extern "C" void kernel_launch(void* const* d_in, const int* in_sizes, int n_in,
                              void* d_out, int out_size, void* d_ws, size_t ws_size,
                              hipStream_t stream)
{
    (void)in_sizes; (void)n_in; (void)out_size; (void)ws_size;
    const float* x   = (const float*)d_in[0];
    const float* W0  = (const float*)d_in[1];
    const float* a0  = (const float*)d_in[2];
    const float* bi0 = (const float*)d_in[3];
    const float* e0  = (const float*)d_in[4];
    const float* g0  = (const float*)d_in[5];
    const float* be0 = (const float*)d_in[6];
    const float* W1  = (const float*)d_in[7];
    const float* a1  = (const float*)d_in[8];
    const float* bi1 = (const float*)d_in[9];
    const float* e1  = (const float*)d_in[10];
    const float* g1  = (const float*)d_in[11];
    const float* be1 = (const float*)d_in[12];
    const float* W2  = (const float*)d_in[13];
    const float* a2  = (const float*)d_in[14];
    const float* bi2 = (const float*)d_in[15];
    const float* e2  = (const float*)d_in[16];
    const float* g2  = (const float*)d_in[17];
    const float* be2 = (const float*)d_in[18];
    const float* hw  = (const float*)d_in[19];
    const float* hb  = (const float*)d_in[20];

    float* ws = (float*)d_ws;
    size_t off = 0;
    float* h0  = ws + off; off += (size_t)BATCH * K0PAD;   // 1024 x 10240
    float* h1  = ws + off; off += (size_t)BATCH * 256;
    float* h2  = ws + off; off += (size_t)BATCH * 128;
    float* Wp0 = ws + off; off += (size_t)256 * K0PAD;
    float* Ap0 = ws + off; off += (size_t)256 * K0PAD;
    float* Wp1 = ws + off; off += (size_t)128 * 256;
    float* Ap1 = ws + off; off += (size_t)128 * 256;
    float* Wp2 = ws + off; off += (size_t)64 * 128;
    float* Ap2 = ws + off; off += (size_t)64 * 128;
    float* rn0 = ws + off; off += BATCH;
    float* rn1 = ws + off; off += BATCH;
    float* rn2 = ws + off; off += BATCH;

    // 1) weight pre-pack (pad K, apply sigmoid to alpha)
    {
        int t0 = 256 * K0PAD;
        prepack_kernel<<<(t0 + 255) / 256, 256, 0, stream>>>(W0, a0, Wp0, Ap0, NFREQ, K0PAD, t0);
        int t1 = 128 * 256;
        prepack_kernel<<<(t1 + 255) / 256, 256, 0, stream>>>(W1, a1, Wp1, Ap1, 256, 256, t1);
        int t2 = 64 * 128;
        prepack_kernel<<<(t2 + 255) / 256, 256, 0, stream>>>(W2, a2, Wp2, Ap2, 128, 128, t2);
    }

    // 2) FFT front-end (160 KB dynamic LDS per workgroup; CDNA5 WGP has 320 KB)
    fft_front_kernel<<<BATCH, FFT_THREADS, 2 * FFT_N * sizeof(float2), stream>>>(x, h0, rn0);

    // 3) fused Hebbian layers (f32 WMMA), head folded into the last one
    hebb_layer_kernel<K0PAD, 256, false><<<BATCH / 16, 512, 0, stream>>>(
        h0, rn0, Wp0, Ap0, bi0, e0, g0, be0, h1, rn1, nullptr, nullptr, nullptr);
    hebb_layer_kernel<256, 128, false><<<BATCH / 16, 256, 0, stream>>>(
        h1, rn1, Wp1, Ap1, bi1, e1, g1, be1, h2, rn2, nullptr, nullptr, nullptr);
    hebb_layer_kernel<128, 64, true><<<BATCH / 16, 128, 0, stream>>>(
        h2, rn2, Wp2, Ap2, bi2, e2, g2, be2, nullptr, nullptr, hw, hb, (float*)d_out);
}